// TGNEncoder_75797582840086
// MI455X (gfx1250) — compile-verified
//
#include <hip/hip_runtime.h>
#include <math.h>

// Problem constants (fixed by the reference).
#define N_NODES 50000
#define N_EDGES 800000
#define IN_CH   100
#define NF_DIM  64
#define IN_TOT  164     // NF_DIM + IN_CH, = 41*4
#define TIME_DIM 100
#define MSG_DIM  64
#define EDGE_DIM 164    // TIME_DIM + MSG_DIM
#define OUT_CH  128
#define HEADS   2
#define HEAD_C  64

typedef float v2f __attribute__((ext_vector_type(2)));
typedef float v8f __attribute__((ext_vector_type(8)));

// Float atomic max via sign-split integer atomics (works with -inf init).
__device__ __forceinline__ void atomic_max_float(float* addr, float v) {
    if (v >= 0.0f) atomicMax((int*)addr, __float_as_int(v));
    else           atomicMin((unsigned int*)addr, (unsigned int)__float_as_int(v));
}

__global__ void tgn_init_kernel(float* __restrict__ amax, float* __restrict__ denom) {
    int i = blockIdx.x * blockDim.x + threadIdx.x;
    if (i < N_NODES * HEADS) { amax[i] = -INFINITY; denom[i] = 0.0f; }
}

// ---------------------------------------------------------------------------
// Kernel 1: node-side GEMMs. One block = 16 node rows. xf = [nf | x] staged in
// LDS; 8 waves each own a 16-col tile and loop over the 4 weight matrices
// (Wq,Wk,Wv,Wskip) with V_WMMA_F32_16X16X4_F32, K = 164 in 41 steps of 4.
// ---------------------------------------------------------------------------
__global__ __launch_bounds__(256)
void tgn_node_gemm_kernel(const float* __restrict__ x, const float* __restrict__ nf,
                          const float* __restrict__ Wq, const float* __restrict__ bq,
                          const float* __restrict__ Wk, const float* __restrict__ bk,
                          const float* __restrict__ Wv, const float* __restrict__ bv,
                          const float* __restrict__ Wskip, const float* __restrict__ bskip,
                          float* __restrict__ qw, float* __restrict__ kw,
                          float* __restrict__ vw, float* __restrict__ out) {
    __shared__ float xf[16][IN_TOT + 4];
    const int tid  = threadIdx.x;
    const int base = blockIdx.x * 16;           // N divisible by 16 -> no guards
    for (int idx = tid; idx < 16 * IN_TOT; idx += 256) {
        int r = idx / IN_TOT, c = idx % IN_TOT;
        int row = base + r;
        xf[r][c] = (c < NF_DIM) ? nf[(size_t)row * NF_DIM + c]
                                : x[(size_t)row * IN_CH + (c - NF_DIM)];
    }
    __syncthreads();

    const int wave = tid >> 5;
    const int lane = tid & 31;
    // A-matrix 16x4 f32 layout: lanes 0-15 -> M=lane, K=0..1; lanes 16-31 -> M=lane-16, K=2..3
    const int mrow = lane & 15;
    const int koff = (lane < 16) ? 0 : 2;
    const int ncol = lane & 15;                 // B/C/D: N = lane % 16
    const float* Wmat[4] = {Wq, Wk, Wv, Wskip};
    const float* bias[4] = {bq, bk, bv, bskip};
    float*       dsts[4] = {qw, kw, vw, out};
    const int col = wave * 16 + ncol;           // column within each 128-wide matrix

    for (int sel = 0; sel < 4; ++sel) {
        const float* W = Wmat[sel];
        v8f acc = {};
        for (int k0 = 0; k0 < IN_TOT; k0 += 4) {
            v2f a, b;
            a.x = xf[mrow][k0 + koff];
            a.y = xf[mrow][k0 + koff + 1];
            b.x = W[(size_t)(k0 + koff)     * OUT_CH + col];
            b.y = W[(size_t)(k0 + koff + 1) * OUT_CH + col];
            acc = __builtin_amdgcn_wmma_f32_16x16x4_f32(false, a, false, b,
                                                        (short)0, acc, false, false);
        }
        const float bb = bias[sel][col];
        float* dst = dsts[sel];
#pragma unroll
        for (int r = 0; r < 8; ++r) {
            int m = (lane < 16) ? r : (8 + r);  // C/D: vgpr r -> M=r / M=8+r
            dst[(size_t)(base + m) * OUT_CH + col] = acc[r] + bb;
        }
    }
}

// ---------------------------------------------------------------------------
// Kernel 2: edge pass. One block = 16 edges. Build edge_attr tile (cos time
// encoding | msg) in LDS, WMMA against We (8 waves x 16 cols), stash e to ws
// and LDS, then compute attention logits + atomic segment-max.
// ---------------------------------------------------------------------------
__global__ __launch_bounds__(256)
void tgn_edge_gemm_kernel(const int* __restrict__ ei, const int* __restrict__ lastu,
                          const int* __restrict__ tt, const float* __restrict__ msg,
                          const float* __restrict__ time_w, const float* __restrict__ time_b,
                          const float* __restrict__ We,
                          const float* __restrict__ qw, const float* __restrict__ kw,
                          float* __restrict__ e_ws, float* __restrict__ alpha_ws,
                          float* __restrict__ amax) {
    __shared__ float ea[16][EDGE_DIM + 4];
    __shared__ float elds[16][OUT_CH];
    __shared__ float red[256];
    const int tid  = threadIdx.x;
    const int base = blockIdx.x * 16;           // E divisible by 16 -> no guards
    for (int idx = tid; idx < 16 * EDGE_DIM; idx += 256) {
        int r = idx / EDGE_DIM, c = idx % EDGE_DIM;
        int eid = base + r;
        float val;
        if (c < TIME_DIM) {
            int s = ei[eid];
            float rel = (float)(lastu[s] - tt[eid]);
            val = cosf(rel * time_w[c] + time_b[c]);
        } else {
            val = msg[(size_t)eid * MSG_DIM + (c - TIME_DIM)];
        }
        ea[r][c] = val;
    }
    __syncthreads();

    const int wave = tid >> 5;
    const int lane = tid & 31;
    const int mrow = lane & 15;
    const int koff = (lane < 16) ? 0 : 2;
    const int col  = wave * 16 + (lane & 15);
    v8f acc = {};
    for (int k0 = 0; k0 < EDGE_DIM; k0 += 4) {
        v2f a, b;
        a.x = ea[mrow][k0 + koff];
        a.y = ea[mrow][k0 + koff + 1];
        b.x = We[(size_t)(k0 + koff)     * OUT_CH + col];
        b.y = We[(size_t)(k0 + koff + 1) * OUT_CH + col];
        acc = __builtin_amdgcn_wmma_f32_16x16x4_f32(false, a, false, b,
                                                    (short)0, acc, false, false);
    }
#pragma unroll
    for (int r = 0; r < 8; ++r) {
        int m = (lane < 16) ? r : (8 + r);
        float val = acc[r];
        elds[m][col] = val;
        e_ws[(size_t)(base + m) * OUT_CH + col] = val;
    }
    __syncthreads();

    // alpha[e,h] = dot(q[dst,h,:], k[src,h,:] + e[e,h,:]) / 8
    // 32 (edge,head) pairs x 8 threads each over the 64-dim head.
    const int pair = tid >> 3;      // 0..31
    const int sub  = tid & 7;
    const int er   = pair >> 1;
    const int h    = pair & 1;
    const int eid  = base + er;
    const int s = ei[eid];
    const int d = ei[N_EDGES + eid];
    float partial = 0.0f;
    for (int c = sub; c < HEAD_C; c += 8) {
        int kc = h * HEAD_C + c;
        partial += qw[(size_t)d * OUT_CH + kc] * (kw[(size_t)s * OUT_CH + kc] + elds[er][kc]);
    }
    red[tid] = partial;
    __syncthreads();
    if (sub == 0) {
        float a2 = 0.0f;
#pragma unroll
        for (int j = 0; j < 8; ++j) a2 += red[(pair << 3) + j];
        float al = a2 * 0.125f;     // 1/sqrt(64)
        alpha_ws[(size_t)eid * HEADS + h] = al;
        atomic_max_float(&amax[(size_t)d * HEADS + h], al);
    }
}

// Kernel 3: ex = exp(alpha - amax[dst]); denom[dst] += ex (alpha overwritten).
__global__ void tgn_softmax_denom_kernel(const int* __restrict__ ei,
                                         const float* __restrict__ amax,
                                         float* __restrict__ alpha_ws,
                                         float* __restrict__ denom) {
    int i = blockIdx.x * blockDim.x + threadIdx.x;
    if (i >= N_EDGES * HEADS) return;
    int e = i >> 1, h = i & 1;
    int d = ei[N_EDGES + e];
    float ex = expf(alpha_ws[i] - amax[d * HEADS + h]);
    alpha_ws[i] = ex;
    atomicAdd(&denom[d * HEADS + h], ex);
}

// Kernel 4: out[dst] += (v[src] + e) * ex / (denom[dst] + 1e-16). Thread per
// (edge, channel); e and v[src] reads are fully coalesced 128-float rows.
__global__ void tgn_aggregate_kernel(const int* __restrict__ ei,
                                     const float* __restrict__ vw,
                                     const float* __restrict__ e_ws,
                                     const float* __restrict__ alpha_ws,
                                     const float* __restrict__ denom,
                                     float* __restrict__ out) {
    size_t i = (size_t)blockIdx.x * blockDim.x + threadIdx.x;   // < E*128 exactly
    int e   = (int)(i >> 7);
    int col = (int)(i & 127);
    int h   = col >> 6;
    int s = ei[e];
    int d = ei[N_EDGES + e];
    float w = alpha_ws[(size_t)e * HEADS + h] / (denom[(size_t)d * HEADS + h] + 1e-16f);
    float val = (vw[(size_t)s * OUT_CH + col] + e_ws[i]) * w;
    atomicAdd(&out[(size_t)d * OUT_CH + col], val);
}

extern "C" void kernel_launch(void* const* d_in, const int* in_sizes, int n_in,
                              void* d_out, int out_size, void* d_ws, size_t ws_size,
                              hipStream_t stream) {
    const float* x      = (const float*)d_in[0];
    const int*   lastu  = (const int*)  d_in[1];
    const int*   ei     = (const int*)  d_in[2];
    const int*   tt     = (const int*)  d_in[3];
    const float* msg    = (const float*)d_in[4];
    const float* nf     = (const float*)d_in[5];
    const float* time_w = (const float*)d_in[6];
    const float* time_b = (const float*)d_in[7];
    const float* Wq     = (const float*)d_in[8];
    const float* bq     = (const float*)d_in[9];
    const float* Wk     = (const float*)d_in[10];
    const float* bk     = (const float*)d_in[11];
    const float* Wv     = (const float*)d_in[12];
    const float* bv     = (const float*)d_in[13];
    const float* We     = (const float*)d_in[14];
    const float* Wskip  = (const float*)d_in[15];
    const float* bskip  = (const float*)d_in[16];
    float* out = (float*)d_out;

    // Workspace layout (floats): q | k | v | amax | denom | alpha | e  (~494 MB)
    float* qw    = (float*)d_ws;
    float* kw    = qw    + (size_t)N_NODES * OUT_CH;
    float* vw    = kw    + (size_t)N_NODES * OUT_CH;
    float* amax  = vw    + (size_t)N_NODES * OUT_CH;
    float* denom = amax  + (size_t)N_NODES * HEADS;
    float* alpha = denom + (size_t)N_NODES * HEADS;
    float* e_ws  = alpha + (size_t)N_EDGES * HEADS;

    tgn_init_kernel<<<(N_NODES * HEADS + 255) / 256, 256, 0, stream>>>(amax, denom);

    tgn_node_gemm_kernel<<<N_NODES / 16, 256, 0, stream>>>(
        x, nf, Wq, bq, Wk, bk, Wv, bv, Wskip, bskip, qw, kw, vw, out);

    tgn_edge_gemm_kernel<<<N_EDGES / 16, 256, 0, stream>>>(
        ei, lastu, tt, msg, time_w, time_b, We, qw, kw, e_ws, alpha, amax);

    tgn_softmax_denom_kernel<<<(N_EDGES * HEADS + 255) / 256, 256, 0, stream>>>(
        ei, amax, alpha, denom);

    tgn_aggregate_kernel<<<(int)(((size_t)N_EDGES * OUT_CH) / 256), 256, 0, stream>>>(
        ei, vw, e_ws, alpha, denom, out);
}